// Gcn_54477365182993
// MI455X (gfx1250) — compile-verified
//
#include <hip/hip_runtime.h>

typedef float v2f __attribute__((ext_vector_type(2)));
typedef float v8f __attribute__((ext_vector_type(8)));

#define N_NODES 100000
#define N_EDGES 1600000
#define F_IN    256
#define DIM     128
#define N_CLS   16
#define M_TILES (N_NODES / 16)   // 6250, exact

__device__ __forceinline__ void atomAddF(float* p, float v) {
    unsafeAtomicAdd(p, v);   // global_atomic_add_f32
}

// ---------------- degree / normalization ----------------
__global__ void k_init_deg(float* __restrict__ deg) {
    int i = blockIdx.x * 256 + threadIdx.x;
    if (i < N_NODES) deg[i] = 1.0f;              // self loop
}

__global__ void k_deg_edges(const int* __restrict__ dst, float* __restrict__ deg) {
    int e = blockIdx.x * 256 + threadIdx.x;
    if (e < N_EDGES) atomAddF(&deg[dst[e]], 1.0f);
}

__global__ void k_dinv(float* __restrict__ d) {
    int i = blockIdx.x * 256 + threadIdx.x;
    if (i < N_NODES) d[i] = rsqrtf(d[i]);        // deg >= 1 always
}

__global__ void k_norm(const int* __restrict__ src, const int* __restrict__ dst,
                       const float* __restrict__ dinv, float* __restrict__ norm) {
    int e = blockIdx.x * 256 + threadIdx.x;
    if (e < N_EDGES) norm[e] = dinv[src[e]] * dinv[dst[e]];
}

// ---------------- GEMM1: h = x @ W1  (f32 WMMA 16x16x4) ----------------
__global__ __launch_bounds__(256) void k_gemm1(const float* __restrict__ x,
                                               const float* __restrict__ W1,
                                               float* __restrict__ h) {
    __shared__ float xs[16 * F_IN];              // 16 KB
    const int row0 = blockIdx.x * 16;
    const int t = threadIdx.x;                   // 0..255 == one column of the tile
    #pragma unroll
    for (int i = 0; i < 16; ++i)
        xs[i * F_IN + t] = x[(size_t)(row0 + i) * F_IN + t];
    __syncthreads();

    const int lane  = t & 31;
    const int wave  = t >> 5;                    // N-tile 0..7
    const int col0  = wave * 16;
    const int ml    = lane & 15;                 // M (for A) / N (for B,C)
    const int khalf = (lane >> 4) * 2;           // lanes 16-31 hold K+2,K+3
    v8f c = {};
    for (int k0 = 0; k0 < F_IN; k0 += 4) {
        const int ka = k0 + khalf;
        v2f a, b;
        a.x = xs[ml * F_IN + ka];
        a.y = xs[ml * F_IN + ka + 1];
        b.x = W1[(size_t)ka       * DIM + col0 + ml];
        b.y = W1[(size_t)(ka + 1) * DIM + col0 + ml];
        c = __builtin_amdgcn_wmma_f32_16x16x4_f32(false, a, false, b,
                                                  (short)0, c, false, false);
    }
    const int mb = (lane >> 4) * 8;              // C: vgpr j -> M = mb + j, N = ml
    #pragma unroll
    for (int j = 0; j < 8; ++j)
        h[(size_t)(row0 + mb + j) * DIM + col0 + ml] = c[j];
}

// ---------------- layer-1 aggregation ----------------
__global__ void k_selfloop1(const float* __restrict__ h, const float* __restrict__ dinv,
                            float* __restrict__ out1) {
    int i = blockIdx.x * 256 + threadIdx.x;      // over N*DIM
    if (i < N_NODES * DIM) {
        float di = dinv[i >> 7];                 // DIM == 128
        out1[i] = h[i] * di * di;
    }
}

__global__ void k_agg1_edges(const float* __restrict__ h, const int* __restrict__ src,
                             const int* __restrict__ dst, const float* __restrict__ norm,
                             float* __restrict__ out1) {
    int gid  = blockIdx.x * 256 + threadIdx.x;   // one wave (32 lanes) per edge
    int e    = gid >> 5;
    int lane = gid & 31;
    if (e >= N_EDGES) return;
    int s = src[e], d = dst[e];
    float w = norm[e];
    const float4 hv = *(const float4*)(h + (size_t)s * DIM + lane * 4);
    float* o = out1 + (size_t)d * DIM + lane * 4;
    atomAddF(o + 0, hv.x * w);
    atomAddF(o + 1, hv.y * w);
    atomAddF(o + 2, hv.z * w);
    atomAddF(o + 3, hv.w * w);
}

__global__ void k_bias_relu(float* __restrict__ h1, const float* __restrict__ b1) {
    int i = blockIdx.x * 256 + threadIdx.x;
    if (i < N_NODES * DIM) {
        float v = h1[i] + b1[i & (DIM - 1)];
        h1[i] = v > 0.0f ? v : 0.0f;
    }
}

// ---------------- GEMM2: h2 = h1 @ W2  (f32 WMMA, N = 16 one tile) ----------------
__global__ __launch_bounds__(128) void k_gemm2(const float* __restrict__ h1,
                                               const float* __restrict__ W2,
                                               float* __restrict__ h2) {
    __shared__ float as[4][16 * DIM];            // 32 KB, one 8KB tile per wave
    const int wave  = threadIdx.x >> 5;
    const int lane  = threadIdx.x & 31;
    const int mtile = blockIdx.x * 4 + wave;
    const bool active = mtile < M_TILES;
    float* sm = as[wave];
    if (active) {
        const float* srcp = h1 + (size_t)mtile * 16 * DIM;   // 2048 contiguous floats
        for (int idx = lane * 4; idx < 16 * DIM; idx += 32 * 4)
            *(float4*)(sm + idx) = *(const float4*)(srcp + idx);
    }
    asm volatile("s_wait_dscnt 0" ::: "memory"); // wave-private LDS: DS in-order + wait
    if (!active) return;

    const int ml    = lane & 15;
    const int khalf = (lane >> 4) * 2;
    v8f c = {};
    #pragma unroll 4
    for (int k0 = 0; k0 < DIM; k0 += 4) {
        const int ka = k0 + khalf;
        v2f a, b;
        a.x = sm[ml * DIM + ka];
        a.y = sm[ml * DIM + ka + 1];
        b.x = W2[ka       * N_CLS + ml];
        b.y = W2[(ka + 1) * N_CLS + ml];
        c = __builtin_amdgcn_wmma_f32_16x16x4_f32(false, a, false, b,
                                                  (short)0, c, false, false);
    }
    const int row0 = mtile * 16;
    const int mb   = (lane >> 4) * 8;
    #pragma unroll
    for (int j = 0; j < 8; ++j)
        h2[(size_t)(row0 + mb + j) * N_CLS + ml] = c[j];
}

// ---------------- layer-2 aggregation + log_softmax ----------------
__global__ void k_selfloop2(const float* __restrict__ h2, const float* __restrict__ dinv,
                            const float* __restrict__ b2, float* __restrict__ out) {
    int i = blockIdx.x * 256 + threadIdx.x;      // over N*16
    if (i < N_NODES * N_CLS) {
        float di = dinv[i >> 4];
        out[i] = h2[i] * di * di + b2[i & 15];
    }
}

__global__ void k_agg2_edges(const float* __restrict__ h2, const int* __restrict__ src,
                             const int* __restrict__ dst, const float* __restrict__ norm,
                             float* __restrict__ out) {
    int gid = blockIdx.x * 256 + threadIdx.x;    // 16 threads per edge
    int e = gid >> 4;
    int c = gid & 15;
    if (e >= N_EDGES) return;
    float w = norm[e];
    atomAddF(out + (size_t)dst[e] * N_CLS + c, h2[(size_t)src[e] * N_CLS + c] * w);
}

__global__ void k_logsoftmax(float* __restrict__ out) {
    int i = blockIdx.x * 256 + threadIdx.x;
    if (i >= N_NODES) return;
    float* row = out + (size_t)i * N_CLS;
    float v[N_CLS];
    float4* p = (float4*)row;
    #pragma unroll
    for (int q = 0; q < 4; ++q) {
        float4 r = p[q];
        v[q * 4 + 0] = r.x; v[q * 4 + 1] = r.y; v[q * 4 + 2] = r.z; v[q * 4 + 3] = r.w;
    }
    float m = v[0];
    #pragma unroll
    for (int c = 1; c < N_CLS; ++c) m = fmaxf(m, v[c]);
    float s = 0.0f;
    #pragma unroll
    for (int c = 0; c < N_CLS; ++c) s += __expf(v[c] - m);
    float l = m + __logf(s);
    #pragma unroll
    for (int q = 0; q < 4; ++q) {
        float4 r;
        r.x = v[q * 4 + 0] - l; r.y = v[q * 4 + 1] - l;
        r.z = v[q * 4 + 2] - l; r.w = v[q * 4 + 3] - l;
        p[q] = r;
    }
}

// ---------------- host launcher ----------------
extern "C" void kernel_launch(void* const* d_in, const int* in_sizes, int n_in,
                              void* d_out, int out_size, void* d_ws, size_t ws_size,
                              hipStream_t stream) {
    const float* x  = (const float*)d_in[0];
    const int*   ei = (const int*)d_in[1];     // [2, E] int32
    const float* W1 = (const float*)d_in[2];
    const float* b1 = (const float*)d_in[3];
    const float* W2 = (const float*)d_in[4];
    const float* b2 = (const float*)d_in[5];
    float* out = (float*)d_out;

    const int* src = ei;
    const int* dst = ei + N_EDGES;

    float* wf   = (float*)d_ws;
    float* dinv = wf;                                       // N (also deg scratch)
    float* norm = dinv + N_NODES;                           // E
    float* h    = norm + N_EDGES;                           // N*DIM
    float* h1   = h    + (size_t)N_NODES * DIM;             // N*DIM
    float* h2   = h1   + (size_t)N_NODES * DIM;             // N*16

    const int B = 256;
    k_init_deg  <<<(N_NODES + B - 1) / B, B, 0, stream>>>(dinv);
    k_deg_edges <<<(N_EDGES + B - 1) / B, B, 0, stream>>>(dst, dinv);
    k_dinv      <<<(N_NODES + B - 1) / B, B, 0, stream>>>(dinv);
    k_norm      <<<(N_EDGES + B - 1) / B, B, 0, stream>>>(src, dst, dinv, norm);

    k_gemm1     <<<M_TILES, 256, 0, stream>>>(x, W1, h);

    k_selfloop1 <<<(N_NODES * DIM + B - 1) / B, B, 0, stream>>>(h, dinv, h1);
    k_agg1_edges<<<((size_t)N_EDGES * 32 + B - 1) / B, B, 0, stream>>>(h, src, dst, norm, h1);
    k_bias_relu <<<(N_NODES * DIM + B - 1) / B, B, 0, stream>>>(h1, b1);

    k_gemm2     <<<(M_TILES + 3) / 4, 128, 0, stream>>>(h1, W2, h2);

    k_selfloop2 <<<(N_NODES * N_CLS + B - 1) / B, B, 0, stream>>>(h2, dinv, b2, out);
    k_agg2_edges<<<((size_t)N_EDGES * N_CLS + B - 1) / B, B, 0, stream>>>(h2, src, dst, norm, out);

    k_logsoftmax<<<(N_NODES + B - 1) / B, B, 0, stream>>>(out);
}